// GraphConvolution_19945828122830
// MI455X (gfx1250) — compile-verified
//
#include <hip/hip_runtime.h>

typedef float v2f __attribute__((ext_vector_type(2)));
typedef float v8f __attribute__((ext_vector_type(8)));

#define DFEAT 64

// ---------------------------------------------------------------------------
// Kernel 1: out[n][d] = bias[d]   (segment_sum base + bias, fused)
// ---------------------------------------------------------------------------
__global__ __launch_bounds__(256) void gcn_init_bias(
    const float* __restrict__ bias, float* __restrict__ out, int total)
{
    int i = blockIdx.x * blockDim.x + threadIdx.x;
    if (i < total) out[i] = bias[i & (DFEAT - 1)];
}

// ---------------------------------------------------------------------------
// Kernel 2: support = X @ W via V_WMMA_F32_16X16X4_F32 (wave32)
// One wave computes a 16x64 tile: 4 accumulators x 16 k-steps = 64 WMMAs.
// A frag (16x4 f32): lane m=lane%16, h=lane/16 -> holds X[m][4ks+2h .. +1]
// B frag (4x16 f32): lane n=lane%16           -> holds W[4ks+2h .. +1][n]
// D frag (16x16 f32): acc[v] = D[v + 8h][lane%16]
// ---------------------------------------------------------------------------
__global__ __launch_bounds__(256) void gcn_gemm_wmma(
    const float* __restrict__ X, const float* __restrict__ W,
    float* __restrict__ S, int nTiles)
{
    __shared__ float WT[DFEAT * DFEAT];   // W transposed: WT[n*64 + k]
    for (int i = threadIdx.x; i < DFEAT * DFEAT; i += 256) {
        int k = i >> 6, n = i & 63;
        WT[n * DFEAT + k] = W[i];
    }
    __syncthreads();

    const int wave = threadIdx.x >> 5;          // 8 waves / block
    const int lane = threadIdx.x & 31;
    const int m    = lane & 15;
    const int h    = lane >> 4;                 // 0 or 1 (K half)
    const int rowTile = blockIdx.x * 8 + wave;  // wave-uniform
    if (rowTile >= nTiles) return;              // EXEC stays all-1s per wave
    const int row0 = rowTile << 4;

    const float* __restrict__ xrow = X + (size_t)(row0 + m) * DFEAT;

    v8f acc0 = {0.f,0.f,0.f,0.f,0.f,0.f,0.f,0.f};
    v8f acc1 = acc0, acc2 = acc0, acc3 = acc0;

    #pragma unroll
    for (int ks = 0; ks < 16; ++ks) {
        const int k = (ks << 2) + (h << 1);
        v2f a  = *(const v2f*)(xrow + k);
        v2f b0 = *(const v2f*)(&WT[( 0 + m) * DFEAT + k]);
        v2f b1 = *(const v2f*)(&WT[(16 + m) * DFEAT + k]);
        v2f b2 = *(const v2f*)(&WT[(32 + m) * DFEAT + k]);
        v2f b3 = *(const v2f*)(&WT[(48 + m) * DFEAT + k]);
        acc0 = __builtin_amdgcn_wmma_f32_16x16x4_f32(false, a, false, b0, (short)0, acc0, false, false);
        acc1 = __builtin_amdgcn_wmma_f32_16x16x4_f32(false, a, false, b1, (short)0, acc1, false, false);
        acc2 = __builtin_amdgcn_wmma_f32_16x16x4_f32(false, a, false, b2, (short)0, acc2, false, false);
        acc3 = __builtin_amdgcn_wmma_f32_16x16x4_f32(false, a, false, b3, (short)0, acc3, false, false);
    }

    float* __restrict__ sbase = S + (size_t)(row0 + (h << 3)) * DFEAT + m;
    #pragma unroll
    for (int v = 0; v < 8; ++v) {
        float* r = sbase + (size_t)v * DFEAT;
        r[ 0] = acc0[v];
        r[16] = acc1[v];
        r[32] = acc2[v];
        r[48] = acc3[v];
    }
}

// Scalar remainder path for row counts not divisible by 16 (unused when
// N % 16 == 0, as here; kept for determinism/correctness on any size).
__global__ __launch_bounds__(64) void gcn_gemm_rem(
    const float* __restrict__ X, const float* __restrict__ W,
    float* __restrict__ S, int startRow, int nrows)
{
    int r = startRow + blockIdx.x;
    int c = threadIdx.x;
    if (r >= nrows) return;
    float acc = 0.f;
    #pragma unroll
    for (int k = 0; k < DFEAT; ++k) acc += X[(size_t)r * DFEAT + k] * W[k * DFEAT + c];
    S[(size_t)r * DFEAT + c] = acc;
}

// ---------------------------------------------------------------------------
// Kernel 3: out[dst] += adj * support[src]; 16 threads/edge, float4 each.
// support + out both fit in the 192MB L2 -> gather/atomics are L2-resident.
// ---------------------------------------------------------------------------
__global__ __launch_bounds__(256) void gcn_scatter(
    const float* __restrict__ S, const float* __restrict__ adj,
    const int* __restrict__ src, const int* __restrict__ dst,
    float* __restrict__ out, int nedges)
{
    long tid = (long)blockIdx.x * blockDim.x + threadIdx.x;
    int e = (int)(tid >> 4);
    int q = (int)(tid & 15);
    if (e >= nedges) return;
    float a = adj[e];
    int s = src[e];
    int d = dst[e];
    const float4 v = *(const float4*)(S + (size_t)s * DFEAT + q * 4);
    float* o = out + (size_t)d * DFEAT + q * 4;
    atomicAdd(o + 0, a * v.x);
    atomicAdd(o + 1, a * v.y);
    atomicAdd(o + 2, a * v.z);
    atomicAdd(o + 3, a * v.w);
}

// ---------------------------------------------------------------------------
extern "C" void kernel_launch(void* const* d_in, const int* in_sizes, int n_in,
                              void* d_out, int out_size, void* d_ws, size_t ws_size,
                              hipStream_t stream)
{
    const float* x    = (const float*)d_in[0];   // [N,64]
    const float* w    = (const float*)d_in[1];   // [64,64]
    const float* bias = (const float*)d_in[2];   // [64]
    const float* adj  = (const float*)d_in[3];   // [E]
    const int*   src  = (const int*)d_in[4];     // [E]
    const int*   dst  = (const int*)d_in[5];     // [E]
    float* out = (float*)d_out;                  // [N,64]
    float* S   = (float*)d_ws;                   // support scratch [N,64]

    const int nnodes = in_sizes[0] / DFEAT;
    const int nedges = in_sizes[3];

    // 1) out = bias (broadcast)
    {
        int total  = out_size;
        int blocks = (total + 255) / 256;
        gcn_init_bias<<<blocks, 256, 0, stream>>>(bias, out, total);
    }

    // 2) S = X @ W  (WMMA over full 16-row tiles, scalar remainder)
    {
        int fullTiles = nnodes / 16;
        if (fullTiles > 0) {
            int blocks = (fullTiles + 7) / 8;
            gcn_gemm_wmma<<<blocks, 256, 0, stream>>>(x, w, S, fullTiles);
        }
        int rem = nnodes - fullTiles * 16;
        if (rem > 0) {
            gcn_gemm_rem<<<rem, 64, 0, stream>>>(x, w, S, fullTiles * 16, nnodes);
        }
    }

    // 3) out[dst] += adj * S[src]
    {
        long threads = (long)nedges * 16;
        int blocks = (int)((threads + 255) / 256);
        gcn_scatter<<<blocks, 256, 0, stream>>>(S, adj, src, dst, out, nedges);
    }
}